// Model_56538949485244
// MI455X (gfx1250) — compile-verified
//
#include <hip/hip_runtime.h>
#include <math.h>

// ---------------------------------------------------------------------------
// e3nn-style 5-layer equivariant GNN for MI455X (gfx1250).
// Radial-MLP input is constant (ones) => per-edge TP weights are per-layer
// constants. Fold radial MLP + CG norms + 1/sqrt(AVG_NEIGH) into a constant
// mix matrix Wbig[128 x NCOLS] per layer, apply it at NODE level as a dense
// fp32 GEMM via V_WMMA_F32_16X16X4_F32 (A tile staged in LDS via
// global_load_async_to_lds_b128), leaving only a light SH contraction +
// atomic scatter per edge.
// ---------------------------------------------------------------------------

#define N_NODES 4096
#define N_EDGES 32768

typedef float v2f __attribute__((ext_vector_type(2)));
typedef float v8f __attribute__((ext_vector_type(8)));

#define INV_SQRT3 0.57735026918962576f
#define INV_SQRT2 0.70710678118654752f

// workspace offsets in floats (total ~16.3 MB)
#define WS_GEOM 0           // E*16
#define WS_X0   524288      // N*128
#define WS_X1   1048576     // N*128
#define WS_Y    1572864     // N*512
#define WS_WBIG 3670016     // 128*512
#define WS_MSG  3735552     // N*128

struct PathDesc { int xbase, mi, mo, dim, ybase, woff; float scale; };

// scales = 1/sqrt(mi*npath[io]) * 1/sqrt(AVG_NEIGH=1.5), folded per path.
__constant__ PathDesc PATHS_MID[14] = {
    {  0, 32, 32, 1,   0,    0, 0.102062073f},  // P0  0e(x0) sh0 -> 32x0e
    { 88,  8, 32, 3,  32, 1024, 0.204124145f},  // P1  1o(x3) sh1 -> 32x0e
    { 32, 32, 32, 1, 128, 1280, 0.102062073f},  // P2  0o(x1) sh0 -> 32x0o
    { 64,  8, 32, 3, 160, 2304, 0.204124145f},  // P3  1e(x2) sh1 -> 32x0o
    {  0, 32, 16, 1, 256, 2560, 0.102062073f},  // P4  0e sh0 -> 16x0e
    { 88,  8, 16, 3, 272, 3072, 0.204124145f},  // P5  1o sh1 -> 16x0e
    { 32, 32,  8, 1, 320, 3200, 0.072168784f},  // P6  0o sh1 -> 8x1e
    { 64,  8,  8, 3, 328, 3456, 0.144337567f},  // P7  1e sh0 -> 8x1e
    { 64,  8,  8, 3, 352, 3520, 0.144337567f},  // P8  1e sh2 -> 8x1e
    { 88,  8,  8, 3, 376, 3584, 0.144337567f},  // P9  1o sh1 -> 8x1e (cross)
    {  0, 32,  8, 1, 400, 3648, 0.072168784f},  // P10 0e sh1 -> 8x1o
    { 64,  8,  8, 3, 408, 3904, 0.144337567f},  // P11 1e sh1 -> 8x1o (cross)
    { 88,  8,  8, 3, 432, 3968, 0.144337567f},  // P12 1o sh0 -> 8x1o
    { 88,  8,  8, 3, 456, 4032, 0.144337567f},  // P13 1o sh2 -> 8x1o
};
__constant__ PathDesc PATHS_L0[3] = {
    {  0, 1, 32, 1,   0,  0, 0.816496581f},     // 0e sh0 -> 32x0e
    {  0, 1, 16, 1, 256, 32, 0.816496581f},     // 0e sh0 -> 16x0e
    {  0, 1,  8, 1, 400, 48, 0.816496581f},     // 0e sh1 -> 8x1o
};
__constant__ PathDesc PATHS_FIN[4] = {
    { 32, 32, 1, 1,  0,   0, 0.102062073f},     // 0o sh0 -> 1x0o
    { 64,  8, 1, 3,  1,  32, 0.204124145f},     // 1e sh1 -> 1x0o
    {  0, 32, 7, 1,  4,  40, 0.102062073f},     // 0e sh0 -> 7x0e
    { 88,  8, 7, 3, 11, 264, 0.204124145f},     // 1o sh1 -> 7x0e
};

__device__ __forceinline__ float silu_f(float x) { return x / (1.0f + expf(-x)); }

// ---------------------------------------------------------------------------
// Per-edge geometry: sh1 = sqrt(3)*n, and M(sh2)[k][i] = sqrt(3/5)*sum_a sh2_a*B5[a][k][i]
// stored as geom[e*16] = { s1x, s1y, s1z, 0, M(3x3 row-major), pad }
// ---------------------------------------------------------------------------
__global__ void edge_geom_kernel(const float* __restrict__ pos,
                                 const int* __restrict__ esrc,
                                 const int* __restrict__ edst,
                                 float* __restrict__ geom) {
    int e = blockIdx.x * blockDim.x + threadIdx.x;
    if (e >= N_EDGES) return;
    int s = esrc[e], d = edst[e];
    float dx = pos[d * 3 + 0] - pos[s * 3 + 0];
    float dy = pos[d * 3 + 1] - pos[s * 3 + 1];
    float dz = pos[d * 3 + 2] - pos[s * 3 + 2];
    float d2 = dx * dx + dy * dy + dz * dz;
    float inv = d2 > 0.0f ? rsqrtf(d2) : 0.0f;
    float nx = dx * inv, ny = dy * inv, nz = dz * inv;
    const float SQ3 = 1.7320508075688772f;
    const float SQ75 = 2.7386127875258306f;              // sqrt(7.5)
    const float S = 0.70710678118654752f;                // 1/sqrt(2)
    const float T = 0.40824829046386302f;                // 1/sqrt(6)
    const float Q = 0.77459666924148338f;                // sqrt(3/5)
    float sh2_0 = SQ75 * 2.0f * S * nx * ny;
    float sh2_1 = SQ75 * 2.0f * S * ny * nz;
    float sh2_2 = SQ75 * 2.0f * S * nx * nz;
    float sh2_3 = SQ75 * S * (nx * nx - ny * ny);
    float sh2_4 = SQ75 * T * (2.0f * nz * nz - nx * nx - ny * ny);
    float M00 = Q * (sh2_3 * S - sh2_4 * T);
    float M11 = Q * (-sh2_3 * S - sh2_4 * T);
    float M22 = Q * (2.0f * T * sh2_4);
    float M01 = Q * S * sh2_0;
    float M12 = Q * S * sh2_1;
    float M02 = Q * S * sh2_2;
    float* g = geom + (size_t)e * 16;
    g[0] = SQ3 * nx; g[1] = SQ3 * ny; g[2] = SQ3 * nz; g[3] = 0.0f;
    g[4] = M00; g[5] = M01; g[6] = M02;
    g[7] = M01; g[8] = M11; g[9] = M12;
    g[10] = M02; g[11] = M12; g[12] = M22;
    g[13] = 0.0f; g[14] = 0.0f; g[15] = 0.0f;
}

__global__ void init_x_kernel(float* __restrict__ X) {
    int idx = blockIdx.x * blockDim.x + threadIdx.x;
    if (idx >= N_NODES * 128) return;
    X[idx] = ((idx & 127) == 0) ? 1.0f : 0.0f;          // 1x0e "ones" input, padded
}

// ---------------------------------------------------------------------------
// Per-layer constant weights: run radial MLP (input = ones) once, expand into
// dense Wbig[128 x ncols] with CG/path norms folded in. Single block.
// ---------------------------------------------------------------------------
__global__ void expand_weights_kernel(const float* __restrict__ w1, const float* __restrict__ b1,
                                      const float* __restrict__ w2, const float* __restrict__ b2,
                                      const float* __restrict__ w3, int nw,
                                      float* __restrict__ Wbig, int ncols,
                                      int which, int npaths) {
    __shared__ float h1[32];
    __shared__ float h2[32];
    __shared__ float wvec[4096];
    int t = threadIdx.x;
    if (t < 32) h1[t] = silu_f(w1[t] + b1[t]);
    __syncthreads();
    if (t < 32) {
        float acc = b2[t];
        for (int k = 0; k < 32; ++k) acc += h1[k] * w2[k * 32 + t];
        h2[t] = silu_f(acc);
    }
    __syncthreads();
    for (int n = t; n < nw; n += blockDim.x) {
        float acc = 0.0f;
        for (int k = 0; k < 32; ++k) acc += h2[k] * w3[k * nw + n];
        wvec[n] = acc;
    }
    for (int i = t; i < 128 * ncols; i += blockDim.x) Wbig[i] = 0.0f;
    __syncthreads();
    const PathDesc* tab = (which == 0) ? PATHS_L0 : (which == 1) ? PATHS_MID : PATHS_FIN;
    for (int p = 0; p < npaths; ++p) {
        PathDesc pd = tab[p];
        int total = pd.mi * pd.mo * pd.dim;
        for (int idx = t; idx < total; idx += blockDim.x) {
            int u = idx / (pd.mo * pd.dim);
            int r = idx - u * pd.mo * pd.dim;
            int v = r / pd.dim;
            int i = r - v * pd.dim;
            Wbig[(size_t)(pd.xbase + u * pd.dim + i) * ncols + pd.ybase + v * pd.dim + i] =
                wvec[pd.woff + u * pd.mo + v] * pd.scale;
        }
    }
}

// ---------------------------------------------------------------------------
// Node-level premix GEMM: Y[N x NCOLS] = X[N x 128] @ Wbig[128 x NCOLS],
// fp32 WMMA 16x16x4. Block = 4 waves over one 16-row tile; the shared
// 16x128 A tile is staged once into LDS with CDNA5 async global->LDS copies
// (ASYNCcnt-tracked), each wave then computes one 16x16 column tile per
// k-step from LDS (A) and global (B, immediate-offset clauses since NCOLS is
// a compile-time constant).
// ---------------------------------------------------------------------------
template <int NCOLS>
__global__ void premix_wmma_kernel(const float* __restrict__ X,
                                   const float* __restrict__ W,
                                   float* __restrict__ Y) {
    __shared__ __align__(16) float shA[16 * 128];        // 8 KB A tile
    int lane = threadIdx.x & 31;
    int wave = threadIdx.x >> 5;
    int row0 = blockIdx.x * 16;
    int col0 = (blockIdx.y * 4 + wave) * 16;

    // ---- async stage A tile: 2048 floats = 512 x b128, 4 per thread -------
    unsigned lds_base = (unsigned)(uintptr_t)(&shA[0]);  // LDS offset = addr[31:0]
#pragma unroll
    for (int i = 0; i < 4; ++i) {
        int flat = threadIdx.x + i * 128;                // b128 index
        int m = flat >> 5;                               // 32 x b128 per row
        int k4 = (flat & 31) << 2;                       // float offset in row
        const float* gsrc = X + (size_t)(row0 + m) * 128 + k4;
        unsigned ldst = lds_base + (unsigned)(flat << 4);
        asm volatile("global_load_async_to_lds_b128 %0, %1, off"
                     :: "v"(ldst), "v"(gsrc)
                     : "memory");
    }
    asm volatile("s_wait_asynccnt 0x0" ::: "memory");
    __syncthreads();                                     // all waves reach barrier

    if (col0 < NCOLS) {                                  // wave-uniform predicate
        int m = lane & 15;
        int kh = lane >> 4;
        const float* arow = &shA[m * 128 + 2 * kh];
        const float* wcol = W + (size_t)(2 * kh) * NCOLS + col0 + m;
        v8f acc = {0.0f, 0.0f, 0.0f, 0.0f, 0.0f, 0.0f, 0.0f, 0.0f};
#pragma unroll 4
        for (int k = 0; k < 128; k += 4) {
            v2f a = *(const v2f*)(arow + k);             // ds_load_b64
            v2f b;
            b.x = wcol[(size_t)k * NCOLS];               // immediate-offset clause
            b.y = wcol[(size_t)(k + 1) * NCOLS];
            acc = __builtin_amdgcn_wmma_f32_16x16x4_f32(false, a, false, b,
                                                        (short)0, acc, false, false);
        }
        float* yb = Y + (size_t)(row0 + 8 * kh) * NCOLS + col0 + m;
#pragma unroll
        for (int r = 0; r < 8; ++r) {
            yb[(size_t)r * NCOLS] = acc[r];              // immediate-offset stores
        }
    }
}

// ---------------------------------------------------------------------------
// Edge conv (hidden layers): one thread per (edge, out-channel c in 0..127).
// Gathers premixed Y[src], contracts with sh1 / M(sh2), atomic-scatters to dst.
// ---------------------------------------------------------------------------
__global__ void edge_conv_hidden_kernel(const float* __restrict__ Y,
                                        const float* __restrict__ geom,
                                        const int* __restrict__ esrc,
                                        const int* __restrict__ edst,
                                        float* __restrict__ msg) {
    int idx = blockIdx.x * blockDim.x + threadIdx.x;
    int e = idx >> 7;
    int c = idx & 127;
    if (e >= N_EDGES) return;
    int src = esrc[e], dst = edst[e];
    const float* y = Y + (size_t)src * 512;
    __builtin_prefetch(y, 0, 0);                                  // global_prefetch_b8
    const float* g = geom + (size_t)e * 16;
    float s1x = g[0], s1y = g[1], s1z = g[2];
    float m;
    if (c < 32) {                                  // 32x0e: P0 + dot(P1, sh1)/sqrt3
        const float* p = y + 32 + c * 3;
        m = y[c] + (p[0] * s1x + p[1] * s1y + p[2] * s1z) * INV_SQRT3;
    } else if (c < 64) {                           // 32x0o: P2 + dot(P3, sh1)/sqrt3
        int v = c - 32;
        const float* p = y + 160 + v * 3;
        m = y[128 + v] + (p[0] * s1x + p[1] * s1y + p[2] * s1z) * INV_SQRT3;
    } else if (c < 80) {                           // 16x0e gates: P4 + dot(P5, sh1)/sqrt3
        int v = c - 64;
        const float* p = y + 272 + v * 3;
        m = y[256 + v] + (p[0] * s1x + p[1] * s1y + p[2] * s1z) * INV_SQRT3;
    } else if (c < 104) {                          // 8x1e: P6*sh1 + P7 + M@P8 + cross(P9,sh1)/sqrt2
        int t = c - 80, v = t / 3, k = t - v * 3;
        float sh1k = (k == 0) ? s1x : (k == 1) ? s1y : s1z;
        const float* Mr = g + 4 + k * 3;
        const float* a8 = y + 352 + v * 3;
        const float* a9 = y + 376 + v * 3;
        float cr = (k == 0) ? a9[1] * s1z - a9[2] * s1y
                 : (k == 1) ? a9[2] * s1x - a9[0] * s1z
                 :            a9[0] * s1y - a9[1] * s1x;
        m = y[320 + v] * sh1k + y[328 + v * 3 + k]
          + Mr[0] * a8[0] + Mr[1] * a8[1] + Mr[2] * a8[2]
          + cr * INV_SQRT2;
    } else {                                       // 8x1o: P10*sh1 + cross(P11)/sqrt2 + P12 + M@P13
        int t = c - 104, v = t / 3, k = t - v * 3;
        float sh1k = (k == 0) ? s1x : (k == 1) ? s1y : s1z;
        const float* Mr = g + 4 + k * 3;
        const float* a11 = y + 408 + v * 3;
        const float* a13 = y + 456 + v * 3;
        float cr = (k == 0) ? a11[1] * s1z - a11[2] * s1y
                 : (k == 1) ? a11[2] * s1x - a11[0] * s1z
                 :            a11[0] * s1y - a11[1] * s1x;
        m = y[400 + v] * sh1k + cr * INV_SQRT2 + y[432 + v * 3 + k]
          + Mr[0] * a13[0] + Mr[1] * a13[1] + Mr[2] * a13[2];
    }
    atomicAdd(&msg[(size_t)dst * 128 + c], m);
}

// ---------------------------------------------------------------------------
// Gate: se->gelu(tanh approx), so->tanh, vectors scaled by sigmoid(gates).
// Writes padded next-layer features (cols 112..127 = 0).
// ---------------------------------------------------------------------------
__global__ void gate_kernel(const float* __restrict__ msg, float* __restrict__ Xn) {
    int idx = blockIdx.x * blockDim.x + threadIdx.x;
    int n = idx >> 7;
    int c = idx & 127;
    if (n >= N_NODES) return;
    const float* m = msg + (size_t)n * 128;
    float out;
    if (c < 32) {
        float x = m[c];
        out = 0.5f * x * (1.0f + tanhf(0.7978845608028654f * (x + 0.044715f * x * x * x)));
    } else if (c < 64) {
        out = tanhf(m[c]);
    } else if (c < 112) {
        int t = c - 64;                                   // 0..47: ve then vo
        float gv = 1.0f / (1.0f + expf(-m[64 + t / 3])); // gates 16x0e
        out = m[80 + t] * gv;
    } else {
        out = 0.0f;
    }
    Xn[(size_t)n * 128 + c] = out;
}

// ---------------------------------------------------------------------------
// Final edge conv: 8 outputs (1x0o + 7x0e), scatter directly into d_out.
// ---------------------------------------------------------------------------
__global__ void edge_conv_final_kernel(const float* __restrict__ Y,
                                       const float* __restrict__ geom,
                                       const int* __restrict__ esrc,
                                       const int* __restrict__ edst,
                                       float* __restrict__ out) {
    int idx = blockIdx.x * blockDim.x + threadIdx.x;
    int e = idx >> 3;
    int c = idx & 7;
    if (e >= N_EDGES) return;
    const float* y = Y + (size_t)esrc[e] * 32;
    const float* g = geom + (size_t)e * 16;
    float s1x = g[0], s1y = g[1], s1z = g[2];
    float m;
    if (c == 0) {
        m = y[0] + (y[1] * s1x + y[2] * s1y + y[3] * s1z) * INV_SQRT3;
    } else {
        int v = c - 1;
        const float* p = y + 11 + v * 3;
        m = y[4 + v] + (p[0] * s1x + p[1] * s1y + p[2] * s1z) * INV_SQRT3;
    }
    atomicAdd(&out[(size_t)edst[e] * 8 + c], m);
}

// ---------------------------------------------------------------------------

extern "C" void kernel_launch(void* const* d_in, const int* in_sizes, int n_in,
                              void* d_out, int out_size, void* d_ws, size_t ws_size,
                              hipStream_t stream) {
    (void)in_sizes; (void)n_in; (void)out_size; (void)ws_size;
    const float* pos = (const float*)d_in[0];
    const int* esrc = (const int*)d_in[1];
    const int* edst = (const int*)d_in[2];
    // params flattened in jax pytree order (dict keys sorted alphabetically):
    // groups: final, layer0, layer1, layer2, layer3; leaves per group: b1,b2,w1,w2,w3
    const float *PB1[5], *PB2[5], *PW1[5], *PW2[5], *PW3[5];
    for (int gi = 0; gi < 5; ++gi) {
        PB1[gi] = (const float*)d_in[3 + 5 * gi + 0];
        PB2[gi] = (const float*)d_in[3 + 5 * gi + 1];
        PW1[gi] = (const float*)d_in[3 + 5 * gi + 2];
        PW2[gi] = (const float*)d_in[3 + 5 * gi + 3];
        PW3[gi] = (const float*)d_in[3 + 5 * gi + 4];
    }
    float* ws = (float*)d_ws;
    float* GEOM = ws + WS_GEOM;
    float* X0 = ws + WS_X0;
    float* X1 = ws + WS_X1;
    float* Yb = ws + WS_Y;
    float* WB = ws + WS_WBIG;
    float* MSG = ws + WS_MSG;

    edge_geom_kernel<<<N_EDGES / 256, 256, 0, stream>>>(pos, esrc, edst, GEOM);
    init_x_kernel<<<(N_NODES * 128) / 256, 256, 0, stream>>>(X0);

    float* Xc = X0;
    float* Xn = X1;
    for (int l = 0; l < 4; ++l) {
        int grp = 1 + l;                         // groups: 0=final, 1..4=layer0..3
        int which = (l == 0) ? 0 : 1;
        int npaths = (l == 0) ? 3 : 14;
        int nw = (l == 0) ? 56 : 4096;
        expand_weights_kernel<<<1, 256, 0, stream>>>(PW1[grp], PB1[grp], PW2[grp], PB2[grp],
                                                     PW3[grp], nw, WB, 512, which, npaths);
        premix_wmma_kernel<512><<<dim3(N_NODES / 16, 8), 128, 0, stream>>>(Xc, WB, Yb);
        hipMemsetAsync(MSG, 0, (size_t)N_NODES * 128 * sizeof(float), stream);
        edge_conv_hidden_kernel<<<(N_EDGES * 128) / 256, 256, 0, stream>>>(Yb, GEOM, esrc, edst, MSG);
        gate_kernel<<<(N_NODES * 128) / 256, 256, 0, stream>>>(MSG, Xn);
        float* tmp = Xc; Xc = Xn; Xn = tmp;
    }
    // final conv -> d_out [N, 8] = [1x0o | 7x0e]
    expand_weights_kernel<<<1, 256, 0, stream>>>(PW1[0], PB1[0], PW2[0], PB2[0], PW3[0],
                                                 320, WB, 32, 2, 4);
    premix_wmma_kernel<32><<<dim3(N_NODES / 16, 1), 128, 0, stream>>>(Xc, WB, Yb);
    hipMemsetAsync(d_out, 0, (size_t)N_NODES * 8 * sizeof(float), stream);
    edge_conv_final_kernel<<<(N_EDGES * 8) / 256, 256, 0, stream>>>(Yb, GEOM, esrc, edst,
                                                                    (float*)d_out);
}